// MoChA_20366734918128
// MI455X (gfx1250) — compile-verified
//
#include <hip/hip_runtime.h>
#include <math.h>

#define BB 128
#define SS 16384
#define DIMM 10
#define NTHREADS 1024
#define WAVES (NTHREADS / 32)          // 32 waves
#define WCHUNK (SS / WAVES)            // 512 positions per wave
#define TILES (WCHUNK / 16)            // 32 tiles of 16 per wave
#define SMEM_BYTES (2 * SS * 4)        // l[] and u[] staged per row

typedef __attribute__((ext_vector_type(2))) float v2f;
typedef __attribute__((ext_vector_type(8))) float v8f;

// ---- fast, branch-free transcendentals (v_exp_f32 / v_log_f32 / v_tanh_f32) ----
__device__ __forceinline__ float fexp(float x) {
  return __builtin_amdgcn_exp2f(x * 1.4426950408889634f);
}
__device__ __forceinline__ float flog(float x) {
  return __builtin_amdgcn_logf(x) * 0.6931471805599453f;
}
__device__ __forceinline__ float fast_tanh(float x) {
#if __has_builtin(__builtin_amdgcn_tanhf)
  return __builtin_amdgcn_tanhf(x);
#else
  float ax = fabsf(x);
  float e = __builtin_amdgcn_exp2f(ax * 2.8853900817779268f);  // exp(2|x|)
  float t = 1.0f - 2.0f * __builtin_amdgcn_rcpf(e + 1.0f);
  return copysignf(t, x);
#endif
}
__device__ __forceinline__ float fsigmoid(float x) {
  return __builtin_amdgcn_rcpf(1.0f + fexp(-x));
}

// ---- DPP cross-lane helpers (pure VALU, no LDS) ----
// ctrl: 0x00-0xFF quad_perm; 0x100+N row_shl:N (shfl_down); 0x110+N row_shr:N
// (shfl_up); 0x140 row_mirror (xor15); 0x141 row_half_mirror (xor7)
template <int CTRL>
__device__ __forceinline__ float dpp_permf(float v) {
  return __int_as_float(
      __builtin_amdgcn_update_dpp(0, __float_as_int(v), CTRL, 0xF, 0xF, true));
}
#define DPP_XOR1 0xB1   // quad_perm [1,0,3,2]
#define DPP_XOR2 0x4E   // quad_perm [2,3,0,1]
#define DPP_HMIR 0x141  // row_half_mirror: xor7 (== xor4 after xor1+xor2)
#define DPP_MIR 0x140   // row_mirror: xor15 (== xor8 after xor1+xor2+xor4)

__device__ __forceinline__ float red_add16(float v) {
  v += dpp_permf<DPP_XOR1>(v);
  v += dpp_permf<DPP_XOR2>(v);
  v += dpp_permf<DPP_HMIR>(v);
  v += dpp_permf<DPP_MIR>(v);
  return v;
}
__device__ __forceinline__ float red_max16(float v) {
  v = fmaxf(v, dpp_permf<DPP_XOR1>(v));
  v = fmaxf(v, dpp_permf<DPP_XOR2>(v));
  v = fmaxf(v, dpp_permf<DPP_HMIR>(v));
  v = fmaxf(v, dpp_permf<DPP_MIR>(v));
  return v;
}
// inclusive prefix-sum across each 16-lane row (classic GCN row_shr scan)
__device__ __forceinline__ float scan_incl16(float v) {
  v += dpp_permf<0x111>(v);
  v += dpp_permf<0x112>(v);
  v += dpp_permf<0x114>(v);
  v += dpp_permf<0x118>(v);
  return v;
}
// group broadcast of lane k (halves are mirrored -> wave readlane is exact)
__device__ __forceinline__ float bcast16(float v, int k) {
  return __int_as_float(__builtin_amdgcn_readlane(__float_as_int(v), k));
}

struct EnergyW {
  v2f b0, b1, b2;  // B-matrix (W^T) K-slices in WMMA B layout
  float am;        // C init for column a = lane&15 : (h @ V^T + bias)[a]
  float wn;        // weight-normed row vector entry wn[a] (0 for a>=10)
  float off;       // vb + r
};

// Energy for 16 in-range sequence positions [base, base+16).
// Returns energy of position base + (lane&15) in every lane (halves mirrored).
__device__ __forceinline__ float energy_tile(const float* __restrict__ encRow,
                                             int base, const EnergyW& W,
                                             int lane) {
  const int m = lane & 15;
  const int half = lane >> 4;
  const float* ep = encRow + (long long)(base + m) * DIMM;

  v8f d;
#pragma unroll
  for (int r = 0; r < 8; ++r) d[r] = W.am;

#pragma unroll
  for (int s = 0; s < 3; ++s) {
    const int kx = 4 * s + 2 * half;          // K index held by this half-wave
    const int kc = (kx + 2 <= DIMM) ? kx : 0; // clamp (half1,s=2 -> dummy in-range)
    float vx = ep[kc + 0];                    // unconditional b64 load
    float vy = ep[kc + 1];
    v2f a;
    a.x = (kx + 0 < DIMM) ? vx : 0.0f;        // v_cndmask, no branches
    a.y = (kx + 1 < DIMM) ? vy : 0.0f;
    v2f b = (s == 0) ? W.b0 : (s == 1) ? W.b1 : W.b2;
    // D(16x16) += A(16x4) x B(4x16)
    d = __builtin_amdgcn_wmma_f32_16x16x4_f32(false, a, false, b, (short)0, d,
                                              false, false);
  }
  // per-column contribution tanh(y)*wn, then all-reduce across the 16 columns
  float acc[8];
#pragma unroll
  for (int r = 0; r < 8; ++r) acc[r] = fast_tanh(d[r]) * W.wn;
#pragma unroll
  for (int r = 0; r < 8; ++r) {
    acc[r] += dpp_permf<DPP_XOR1>(acc[r]);
    acc[r] += dpp_permf<DPP_XOR2>(acc[r]);
    acc[r] += dpp_permf<DPP_HMIR>(acc[r]);
    acc[r] += dpp_permf<DPP_MIR>(acc[r]);
  }
  // acc[r] now = row (r + 8*half) sum, duplicated across all lanes of the half.
  // Each lane wants row m: mux acc[m&7], then swap halves if m>>3 != half.
  const int r3 = m & 7;
  float s01 = (r3 & 1) ? acc[1] : acc[0];
  float s23 = (r3 & 1) ? acc[3] : acc[2];
  float s45 = (r3 & 1) ? acc[5] : acc[4];
  float s67 = (r3 & 1) ? acc[7] : acc[6];
  float s0123 = (r3 & 2) ? s23 : s01;
  float s4567 = (r3 & 2) ? s67 : s45;
  float sel = (r3 & 4) ? s4567 : s0123;
  float other = __shfl_xor(sel, 16);  // cross the two 16-lane halves
  float eraw = ((m >> 3) != half) ? other : sel;
  return eraw + W.off;
}

extern __shared__ float g_smem[];  // [SS] log(1-p), [SS] u

__global__ __launch_bounds__(NTHREADS) void mocha_kernel(
    const float* __restrict__ enc, const float* __restrict__ h,
    const float* __restrict__ pa, const float* __restrict__ noise,
    const float* __restrict__ Wm, const float* __restrict__ Vm,
    const float* __restrict__ bm, const float* __restrict__ vwm,
    const float* __restrict__ vgm, const float* __restrict__ vbm,
    const float* __restrict__ rm, const float* __restrict__ Wc,
    const float* __restrict__ Vc, const float* __restrict__ bc,
    const float* __restrict__ vwc, const float* __restrict__ vgc,
    const float* __restrict__ vbc, const float* __restrict__ rc,
    float* __restrict__ outAlpha, float* __restrict__ outBeta) {
  float* sLog = g_smem;       // log(clip(1-p)) for all S positions of this row
  float* sUv = g_smem + SS;   // chunk energy u for all S positions

  __shared__ float sWm[DIMM * DIMM], sWc[DIMM * DIMM];
  __shared__ float sAm[16], sAc[16], sWnm[16], sWnc[16], sOff[2];
  __shared__ float sLs[WAVES], sUs[WAVES], sTs[WAVES];

  const int tid = threadIdx.x;
  const int row = blockIdx.x;
  const int lane = tid & 31;
  const int w = tid >> 5;
  const int m = lane & 15;
  const int half = lane >> 4;

  if (tid < DIMM * DIMM) { sWm[tid] = Wm[tid]; sWc[tid] = Wc[tid]; }
  if (tid < 16) {
    float am = 0.f, ac = 0.f;
    if (tid < DIMM) {
      am = bm[tid]; ac = bc[tid];
      for (int dd = 0; dd < DIMM; ++dd) {
        float hv = h[row * DIMM + dd];
        am += hv * Vm[tid * DIMM + dd];
        ac += hv * Vc[tid * DIMM + dd];
      }
    }
    sAm[tid] = am; sAc[tid] = ac;
  }
  if (tid == 0) {
    float nm = 0.f, nc = 0.f;
    for (int e = 0; e < DIMM; ++e) { nm += vwm[e] * vwm[e]; nc += vwc[e] * vwc[e]; }
    nm = sqrtf(nm); nc = sqrtf(nc);
    for (int a = 0; a < 16; ++a) {
      sWnm[a] = (a < DIMM) ? vgm[0] * vwm[a] / nm : 0.f;
      sWnc[a] = (a < DIMM) ? vgc[0] * vwc[a] / nc : 0.f;
    }
    sOff[0] = vbm[0] + rm[0];
    sOff[1] = vbc[0] + rc[0];
  }
  __syncthreads();

  EnergyW Em, Ec;
  Em.am = sAm[m]; Em.wn = sWnm[m]; Em.off = sOff[0];
  Ec.am = sAc[m]; Ec.wn = sWnc[m]; Ec.off = sOff[1];
#pragma unroll
  for (int s = 0; s < 3; ++s) {
    const int kx = 4 * s + 2 * half;
    float bxm = 0.f, bym = 0.f, bxc = 0.f, byc = 0.f;
    if (m < DIMM) {
      if (kx + 0 < DIMM) { bxm = sWm[m * DIMM + kx];     bxc = sWc[m * DIMM + kx]; }
      if (kx + 1 < DIMM) { bym = sWm[m * DIMM + kx + 1]; byc = sWc[m * DIMM + kx + 1]; }
    }
    v2f bmv; bmv.x = bxm; bmv.y = bym;
    v2f bcv; bcv.x = bxc; bcv.y = byc;
    if (s == 0)      { Em.b0 = bmv; Ec.b0 = bcv; }
    else if (s == 1) { Em.b1 = bmv; Ec.b1 = bcv; }
    else             { Em.b2 = bmv; Ec.b2 = bcv; }
  }

  const float* encRow = enc + (long long)row * SS * DIMM;
  const float* noiseRow = noise + (long long)row * SS;
  const float* paRow = pa + (long long)row * SS;
  const int waveBase = w * WCHUNK;
  float* aRow = outAlpha + (long long)row * SS;
  float* bRow = outBeta + (long long)row * SS;

  // ---- Phase A: the ONLY energy pass. Stage l = log(1-p) and u in LDS. ----
  float lsum = 0.f, um = -3.4e38f;
  for (int t = 0; t < TILES; ++t) {
    const int base = waveBase + 16 * t;
    const int pos = base + m;
    float em = energy_tile(encRow, base, Em, lane);
    float uc = energy_tile(encRow, base, Ec, lane);
    float p = fsigmoid(em + noiseRow[pos]);
    float l = flog(fminf(fmaxf(1.f - p, 1e-10f), 1.f));
    if (lane < 16) { sLog[pos] = l; sUv[pos] = uc; }
    lsum += l;
    um = fmaxf(um, uc);
  }
  lsum = red_add16(lsum);
  um = red_max16(um);
  if (lane == 0) { sLs[w] = lsum; sUs[w] = um; }
  __syncthreads();  // also publishes sLog/sUv block-wide
  float lseed = 0.f, umax = -3.4e38f;
  for (int j = 0; j < WAVES; ++j) {
    if (j < w) lseed += sLs[j];
    umax = fmaxf(umax, sUs[j]);
  }

  // ---- Phase B: per-wave sum of t = pa * exp(-cumsum_l) (pure LDS/scan) ----
  float cCarry = lseed, tsum = 0.f;
  for (int t = 0; t < TILES; ++t) {
    const int pos = waveBase + 16 * t + m;
    float l = sLog[pos];
    float incl = scan_incl16(l);
    float cv = cCarry + incl;
    tsum += paRow[pos] * fexp(-cv);
    cCarry += bcast16(incl, 15);
  }
  tsum = red_add16(tsum);
  if (lane == 0) sTs[w] = tsum;
  __syncthreads();
  float Tseed = 0.f;
  for (int j = 0; j < w; ++j) Tseed += sTs[j];

  // ---- Phase C: alpha, exp_u, width-3 moving sums -> beta ----
  cCarry = lseed;
  float TCarry = Tseed;
  // exp_u just left of this wave's chunk (zero-padded at row start)
  float ep1 = (waveBase > 0) ? fmaxf(fexp(sUv[waveBase - 1] - umax), 1e-5f) : 0.f;
  float ep2 = (waveBase > 1) ? fmaxf(fexp(sUv[waveBase - 2] - umax), 1e-5f) : 0.f;
  float ce14 = 0.f, ce15 = 0.f, cr14 = 0.f, cr15 = 0.f;
  int prevBase = 0;
  bool havePrev = false;
  for (int t = 0; t <= TILES; ++t) {  // t==TILES is the ghost tile
    const int base = waveBase + 16 * t;
    const int pos = base + m;
    const bool pv = pos < SS;
    const int cp = pv ? pos : (SS - 1);     // clamped, unconditional loads
    float l = sLog[cp];
    float uv = sUv[cp];
    float pav = paRow[cp];
    float p = 1.f - fexp(l);                // reconstruct p_select from l
    float inclL = scan_incl16(l);
    float cv = cCarry + inclL;
    float tv = pv ? pav * fexp(-cv) : 0.f;
    float inclT = scan_incl16(tv);
    float Tv = TCarry + inclT;
    float alpha = pv ? p * fexp(cv) * Tv : 0.f;
    float e = pv ? fmaxf(fexp(uv - umax), 1e-5f) : 0.f;
    // denom_i = e_{i-2} + e_{i-1} + e_i (zero-padded at row start)
    float s1 = dpp_permf<0x111>(e);         // shfl_up 1 (zero-fill)
    float s2 = dpp_permf<0x112>(e);         // shfl_up 2 (zero-fill)
    float em1 = (m >= 1) ? s1 : ep1;
    float em2 = (m >= 2) ? s2 : ((m == 1) ? ep1 : ep2);
    float denom = e + em1 + em2;
    float ratio = pv ? alpha / denom : 0.f;
    float rp1 = dpp_permf<0x101>(ratio);    // shfl_down 1
    float rp2 = dpp_permf<0x102>(ratio);    // shfl_down 2
    float r0 = bcast16(ratio, 0);
    float r1 = bcast16(ratio, 1);
    if (lane < 16) {
      if (t < TILES) {
        aRow[pos] = alpha;
        if (m < 14) bRow[pos] = e * (ratio + rp1 + rp2);
      }
      if (havePrev) {  // finish previous tile's last two beta values
        if (m == 14) bRow[prevBase + 14] = ce14 * (cr14 + cr15 + r0);
        if (m == 15) bRow[prevBase + 15] = ce15 * (cr15 + r0 + r1);
      }
    }
    cCarry += bcast16(inclL, 15);
    TCarry += bcast16(inclT, 15);
    ce14 = bcast16(e, 14);
    ce15 = bcast16(e, 15);
    cr14 = bcast16(ratio, 14);
    cr15 = bcast16(ratio, 15);
    ep2 = ce14;
    ep1 = ce15;
    prevBase = base;
    havePrev = (t < TILES);
  }
}

extern "C" void kernel_launch(void* const* d_in, const int* in_sizes, int n_in,
                              void* d_out, int out_size, void* d_ws,
                              size_t ws_size, hipStream_t stream) {
  (void)in_sizes; (void)n_in; (void)out_size; (void)d_ws; (void)ws_size;
  const float* enc   = (const float*)d_in[0];
  const float* h     = (const float*)d_in[1];
  const float* pa    = (const float*)d_in[2];
  const float* noise = (const float*)d_in[3];
  const float* Wm  = (const float*)d_in[4];
  const float* Vm  = (const float*)d_in[5];
  const float* bm  = (const float*)d_in[6];
  const float* vwm = (const float*)d_in[7];
  const float* vgm = (const float*)d_in[8];
  const float* vbm = (const float*)d_in[9];
  const float* rm  = (const float*)d_in[10];
  const float* Wc  = (const float*)d_in[11];
  const float* Vc  = (const float*)d_in[12];
  const float* bc  = (const float*)d_in[13];
  const float* vwc = (const float*)d_in[14];
  const float* vgc = (const float*)d_in[15];
  const float* vbc = (const float*)d_in[16];
  const float* rc  = (const float*)d_in[17];
  float* outAlpha = (float*)d_out;
  float* outBeta  = outAlpha + (long long)BB * SS;

  mocha_kernel<<<dim3(BB), dim3(NTHREADS), SMEM_BYTES, stream>>>(
      enc, h, pa, noise, Wm, Vm, bm, vwm, vgm, vbm, rm, Wc, Vc, bc, vwc, vgc,
      vbc, rc, outAlpha, outBeta);
}